// BestRQModel_78812649881913
// MI455X (gfx1250) — compile-verified
//
#include <hip/hip_runtime.h>
#include <hip/hip_bf16.h>
#include <math.h>
#include <stdint.h>

typedef __attribute__((ext_vector_type(2))) float v2f;
typedef __attribute__((ext_vector_type(8))) float v8f;

#define M_TOTAL 16384   // B*T = 8*2048
#define F_IN    320
#define D_DIM   16
#define K_CB    8192
#define KC      512     // codewords staged per LDS chunk (32 KB)
#define NCHUNK  (K_CB / KC)
#define EPSN    1e-12f

// ---------------------------------------------------------------------------
// Kernel 1: proj = x @ P, then L2-normalize rows of proj (D=16).
// Block = 256 threads handles 16 rows; 16 threads per row (one per d).
// ---------------------------------------------------------------------------
__global__ __launch_bounds__(256) void proj_norm_kernel(
    const float* __restrict__ x, const float* __restrict__ P,
    float* __restrict__ projn) {
  __shared__ float sP[F_IN * D_DIM];   // 20 KB, projection [F][D]
  __shared__ float sX[16 * F_IN];      // 20 KB, 16 rows of x
  const int tid = threadIdx.x;
  const int mBase = blockIdx.x * 16;

  const float4* gP = (const float4*)P;
  const float4* gX = (const float4*)(x + (size_t)mBase * F_IN);
  float4* lP = (float4*)sP;
  float4* lX = (float4*)sX;
  for (int i = tid; i < (F_IN * D_DIM) / 4; i += 256) {  // 1280 float4 each
    lP[i] = gP[i];
    lX[i] = gX[i];
  }
  __syncthreads();

  const int row = tid >> 4;
  const int d   = tid & 15;
  const float* xr = sX + row * F_IN;
  float acc = 0.f;
#pragma unroll 4
  for (int f = 0; f < F_IN; ++f) acc = fmaf(xr[f], sP[f * D_DIM + d], acc);

  // sum of squares across the 16 lanes of this row (width-16 groups align
  // with rows: each wave32 holds rows {2w, 2w+1})
  float s = acc * acc;
  s += __shfl_xor(s, 8, 16);
  s += __shfl_xor(s, 4, 16);
  s += __shfl_xor(s, 2, 16);
  s += __shfl_xor(s, 1, 16);
  float out = acc / fmaxf(sqrtf(s), EPSN);
  projn[(size_t)(mBase + row) * D_DIM + d] = out;  // fully coalesced
}

// ---------------------------------------------------------------------------
// Kernel 2: L2-normalize codebook rows [K=8192, D=16] into workspace.
// ---------------------------------------------------------------------------
__global__ __launch_bounds__(256) void cb_norm_kernel(
    const float* __restrict__ cb, float* __restrict__ cbn) {
  const int g = blockIdx.x * 256 + threadIdx.x;  // == row*16 + d
  float v = cb[g];
  float s = v * v;
  s += __shfl_xor(s, 8, 16);
  s += __shfl_xor(s, 4, 16);
  s += __shfl_xor(s, 2, 16);
  s += __shfl_xor(s, 1, 16);
  cbn[g] = v / fmaxf(sqrtf(s), EPSN);
}

// ---------------------------------------------------------------------------
// Async memory->LDS staging of one 32 KB codebook chunk (2048 x b128,
// 8 instructions per wave). Tracked by ASYNCcnt, no VGPR round-trip.
// ---------------------------------------------------------------------------
__device__ __forceinline__ void async_stage_chunk(
    const float* __restrict__ gsrc, float* lds_dst, int tid) {
  // Low 32 bits of a generic shared-memory address == LDS byte offset
  // (shared aperture mapping).
  uint32_t lbase = (uint32_t)(size_t)lds_dst;
#pragma unroll
  for (int j = 0; j < 8; ++j) {
    const int idx = tid + j * 256;  // float4 index within chunk
    uint64_t ga = (uint64_t)(size_t)(gsrc + idx * 4);
    uint32_t la = lbase + (uint32_t)idx * 16u;
    asm volatile("global_load_async_to_lds_b128 %0, %1, off"
                 :: "v"(la), "v"(ga)
                 : "memory");
  }
}

// ---------------------------------------------------------------------------
// Kernel 3: sims = proj_n @ cb_n^T via V_WMMA_F32_16X16X4_F32 (4 chained
// WMMAs cover K=16), fused running argmax over all 8192 codewords.
// Block = 256 threads = 8 waves; each wave owns one 16-row M tile.
// Codebook chunks double-buffered in LDS via async global->LDS loads so the
// next 32 KB chunk streams in while the current one feeds the WMMAs.
// ---------------------------------------------------------------------------
__global__ __launch_bounds__(256) void argmax_kernel(
    const float* __restrict__ projn, const float* __restrict__ cbn,
    float* __restrict__ targets) {
  __shared__ float sCB[2][KC * D_DIM];  // 2 x 32 KB codebook chunks
  __shared__ float sRV[8][16][16];      // 8 KB per-wave best values
  __shared__ int   sRI[8][16][16];      // 8 KB per-wave best indices

  const int tid  = threadIdx.x;
  const int lane = tid & 31;
  const int wave = tid >> 5;
  const int hi   = lane >> 4;   // which half of the wave
  const int lo   = lane & 15;   // m for A-loads, n (codeword column) for B/D
  const int mBase = blockIdx.x * 128 + wave * 16;

  // --- Load A tile (proj_n rows mBase..mBase+15) per WMMA K-chunk layout ---
  // A 16x4 f32: lane(lo,hi) holds A[lo][4i+2hi], A[lo][4i+2hi+1]
  v2f a[4];
  {
    const float* ar = projn + (size_t)(mBase + lo) * D_DIM;
#pragma unroll
    for (int i = 0; i < 4; ++i) {
      a[i].x = ar[4 * i + 2 * hi];
      a[i].y = ar[4 * i + 2 * hi + 1];
    }
  }

  float bestV[8];
  int   bestK[8];
#pragma unroll
  for (int r = 0; r < 8; ++r) { bestV[r] = -3.0e38f; bestK[r] = 0; }

  // Prime the pipeline: stage chunk 0 into buffer 0.
  async_stage_chunk(cbn, &sCB[0][0], tid);

  for (int c = 0; c < NCHUNK; ++c) {
    // Stage next chunk into the other buffer while we compute this one.
    if (c + 1 < NCHUNK) {
      async_stage_chunk(cbn + (size_t)(c + 1) * KC * D_DIM,
                        &sCB[(c + 1) & 1][0], tid);
      // 16 outstanding, wait until current chunk's 8 are done (in-order).
      asm volatile("s_wait_asynccnt 0x8" ::: "memory");
    } else {
      asm volatile("s_wait_asynccnt 0x0" ::: "memory");
    }
    __syncthreads();  // all waves' async data visible in LDS

    const float* buf = &sCB[c & 1][0];
    const int kBase = c * KC;
#pragma unroll 2
    for (int kt = 0; kt < KC / 16; ++kt) {
      // B 4x16: lane(lo=n,hi) holds cbn[k0+n][4i+2hi], cbn[k0+n][4i+2hi+1]
      const float* br = buf + (kt * 16 + lo) * D_DIM;
      v8f cacc = {};
#pragma unroll
      for (int i = 0; i < 4; ++i) {
        v2f b;
        b.x = br[4 * i + 2 * hi];
        b.y = br[4 * i + 2 * hi + 1];
        cacc = __builtin_amdgcn_wmma_f32_16x16x4_f32(
            /*neg_a=*/false, a[i], /*neg_b=*/false, b,
            /*c_mod=*/(short)0, cacc, /*reuse_a=*/false, /*reuse_b=*/false);
      }
      // D layout: VGPR r -> (M = r + 8*hi, N = lo); same codeword per lane
      const int kIdx = kBase + kt * 16 + lo;
#pragma unroll
      for (int r = 0; r < 8; ++r) {
        if (cacc[r] > bestV[r]) { bestV[r] = cacc[r]; bestK[r] = kIdx; }
      }
    }
    __syncthreads();  // done reading this buffer before it is overwritten
  }

  // --- Final per-row argmax across the 16 codeword columns ---
#pragma unroll
  for (int r = 0; r < 8; ++r) {
    sRV[wave][r + 8 * hi][lo] = bestV[r];
    sRI[wave][r + 8 * hi][lo] = bestK[r];
  }
  __syncthreads();
  if (lane < 16) {
    float bv = sRV[wave][lane][0];
    int   bk = sRI[wave][lane][0];
#pragma unroll
    for (int n = 1; n < 16; ++n) {
      float v = sRV[wave][lane][n];
      int   k = sRI[wave][lane][n];
      // argmax of sims == argmin of 2-2*sims; ties -> lowest index
      if (v > bv || (v == bv && k < bk)) { bv = v; bk = k; }
    }
    targets[mBase + lane] = (float)bk;
  }
}

// ---------------------------------------------------------------------------
extern "C" void kernel_launch(void* const* d_in, const int* in_sizes, int n_in,
                              void* d_out, int out_size, void* d_ws,
                              size_t ws_size, hipStream_t stream) {
  const float* x  = (const float*)d_in[0];  // [8,2048,320]
  const float* P  = (const float*)d_in[1];  // [320,16]
  const float* cb = (const float*)d_in[2];  // [8192,16]

  float* out     = (float*)d_out;
  float* targets = out;                // 16384 (indices stored as floats)
  float* projn   = out + M_TOTAL;      // 16384*16
  float* cbn     = (float*)d_ws;       // 8192*16 floats = 512 KB scratch

  proj_norm_kernel<<<M_TOTAL / 16, 256, 0, stream>>>(x, P, projn);
  cb_norm_kernel<<<(K_CB * D_DIM) / 256, 256, 0, stream>>>(cb, cbn);
  argmax_kernel<<<M_TOTAL / 128, 256, 0, stream>>>(projn, cbn, targets);
}